// KVCache_53523882442922
// MI455X (gfx1250) — compile-verified
//
#include <hip/hip_runtime.h>
#include <stdint.h>

// ---------------------------------------------------------------------------
// KV-cache single-token int8 quantize + scatter, plus full-cache emission.
//   B=8, S=4096, H=16, D=128.  Output (float32, concatenated):
//     [0]        ck  (S,H,B,D)  = 67,108,864
//     [67108864] cks (S,H,B,1)  =    524,288
//     [67633152] cv  (S,H,B,D)  = 67,108,864
//     [134742016]cvs (S,H,B,1)  =    524,288
// Dominant cost: ~1.08 GB of streaming traffic -> NT-hinted B128 copy kernels.
// CDNA5 paths used: global_load_async_to_lds_b128 (+ s_wait_asynccnt),
// NT cache hints, wave32 shuffle reduction.
// ---------------------------------------------------------------------------

typedef int   v4i __attribute__((ext_vector_type(4)));
typedef float v4f __attribute__((ext_vector_type(4)));

#define B_ 8
#define S_ 4096
#define H_ 16
#define D_ 128

static const long O_CK  = 0L;
static const long O_CKS = 67108864L;
static const long O_CV  = 67633152L;
static const long O_CVS = 134742016L;

// Streaming int32 -> float32 convert-copy, non-temporal both sides, B128.
__global__ void kv_copy_i32_f32_nt(const int* __restrict__ src,
                                   float* __restrict__ dst, long n4) {
  long i      = (long)blockIdx.x * blockDim.x + threadIdx.x;
  long stride = (long)gridDim.x * blockDim.x;
  const v4i* s = (const v4i*)src;
  v4f*       d = (v4f*)dst;
  for (; i < n4; i += stride) {
    v4i v = __builtin_nontemporal_load(&s[i]);
    v4f f = __builtin_convertvector(v, v4f);
    __builtin_nontemporal_store(f, &d[i]);
  }
}

// Streaming float32 copy, non-temporal, B128.
__global__ void kv_copy_f32_nt(const float* __restrict__ src,
                               float* __restrict__ dst, long n4) {
  long i      = (long)blockIdx.x * blockDim.x + threadIdx.x;
  long stride = (long)gridDim.x * blockDim.x;
  const v4f* s = (const v4f*)src;
  v4f*       d = (v4f*)dst;
  for (; i < n4; i += stride) {
    v4f v = __builtin_nontemporal_load(&s[i]);
    __builtin_nontemporal_store(v, &d[i]);
  }
}

// Quantize the new token's K/V rows and scatter into the caches at position p.
// 16 blocks x 256 threads; one wave32 per (h,b) row for K and V.
// New-token rows are staged into LDS via the CDNA5 async-to-LDS path.
__global__ void kv_quant_update(const float* __restrict__ new_key,
                                const float* __restrict__ new_value,
                                const int* __restrict__ posp,
                                float* __restrict__ out) {
  __shared__ float lds[2048];            // [0..1023] key rows, [1024..2047] value rows

  const int t    = threadIdx.x;          // 0..255
  const int wave = t >> 5;               // 0..7 : local row
  const int lane = t & 31;
  const int R    = blockIdx.x * 8 + wave; // global row 0..127 over (h,b)
  const int h    = R >> 3;               // R / B_
  const int b    = R & 7;                // R % B_

  // source element offset in (B,1,H,D) layout: new[b,0,h, lane*4 .. lane*4+3]
  const int src_elem = (b * H_ + h) * D_ + lane * 4;
  const int voff     = src_elem * 4;     // byte offset for GVS addressing

  // LDS byte addresses (generic-pointer low 32 bits == LDS offset)
  uint32_t lds_k = (uint32_t)(uintptr_t)&lds[wave * D_ + lane * 4];
  uint32_t lds_v = (uint32_t)(uintptr_t)&lds[1024 + wave * D_ + lane * 4];

  // Async DMA 16B per lane into LDS (ASYNCcnt-tracked), then wait.
  asm volatile("global_load_async_to_lds_b128 %0, %1, %2 offset:0"
               :: "v"(lds_k), "v"(voff), "s"(new_key) : "memory");
  asm volatile("global_load_async_to_lds_b128 %0, %1, %2 offset:0"
               :: "v"(lds_v), "v"(voff), "s"(new_value) : "memory");
  asm volatile("s_wait_asynccnt 0" ::: "memory");

  v4f xk = *(const v4f*)&lds[wave * D_ + lane * 4];
  v4f xv = *(const v4f*)&lds[1024 + wave * D_ + lane * 4];

  // per-lane abs-max, then wave32 reduction (5 xor-shuffle steps)
  float mk = fmaxf(fmaxf(fabsf(xk.x), fabsf(xk.y)), fmaxf(fabsf(xk.z), fabsf(xk.w)));
  float mv = fmaxf(fmaxf(fabsf(xv.x), fabsf(xv.y)), fmaxf(fabsf(xv.z), fabsf(xv.w)));
  #pragma unroll
  for (int o = 16; o > 0; o >>= 1) {
    mk = fmaxf(mk, __shfl_xor(mk, o, 32));
    mv = fmaxf(mv, __shfl_xor(mv, o, 32));
  }

  const int p = posp[0];
  const float rk = 127.5f / mk;
  const float rv = 127.5f / mv;

  // jnp.rint(x).astype(int8): round-half-even, then int8 wraparound; emit as f32.
  #define Q8(x) ((float)(signed char)(int)rintf(x))
  v4f qk, qv;
  qk.x = Q8(xk.x * rk); qk.y = Q8(xk.y * rk); qk.z = Q8(xk.z * rk); qk.w = Q8(xk.w * rk);
  qv.x = Q8(xv.x * rv); qv.y = Q8(xv.y * rv); qv.z = Q8(xv.z * rv); qv.w = Q8(xv.w * rv);
  #undef Q8

  const long row = ((long)p * H_ + h) * B_ + b;   // row index in (S,H,B,*) caches
  *(v4f*)(out + O_CK + row * D_ + lane * 4) = qk;
  *(v4f*)(out + O_CV + row * D_ + lane * 4) = qv;
  if (lane == 0) {
    out[O_CKS + row] = mk;
    out[O_CVS + row] = mv;
  }
}

extern "C" void kernel_launch(void* const* d_in, const int* in_sizes, int n_in,
                              void* d_out, int out_size, void* d_ws, size_t ws_size,
                              hipStream_t stream) {
  const float* new_key   = (const float*)d_in[0];  // (8,1,16,128) f32
  const float* new_value = (const float*)d_in[1];  // (8,1,16,128) f32
  const int*   ck        = (const int*)  d_in[2];  // (4096,16,8,128) int
  const float* cks       = (const float*)d_in[3];  // (4096,16,8,1) f32
  const int*   cv        = (const int*)  d_in[4];  // (4096,16,8,128) int
  const float* cvs       = (const float*)d_in[5];  // (4096,16,8,1) f32
  const int*   pos       = (const int*)  d_in[6];  // scalar
  float* out = (float*)d_out;

  const long nKV4 = 67108864L / 4;   // 16,777,216 x B128
  const long nS4  = 524288L / 4;     // 131,072  x B128

  // Bulk emission of the (updated-everywhere-but-row-p) caches, NT streaming.
  kv_copy_i32_f32_nt<<<4096, 256, 0, stream>>>(ck,  out + O_CK,  nKV4);
  kv_copy_f32_nt    <<<256,  256, 0, stream>>>(cks, out + O_CKS, nS4);
  kv_copy_i32_f32_nt<<<4096, 256, 0, stream>>>(cv,  out + O_CV,  nKV4);
  kv_copy_f32_nt    <<<256,  256, 0, stream>>>(cvs, out + O_CVS, nS4);

  // Quantize + scatter the new token at position p (overwrites row p in order).
  kv_quant_update<<<16, 256, 0, stream>>>(new_key, new_value, pos, out);
}